// SimilarityConsistencyLoss_61993557951064
// MI455X (gfx1250) — compile-verified
//
#include <hip/hip_runtime.h>
#include <math.h>

typedef float v2f __attribute__((ext_vector_type(2)));
typedef float v8f __attribute__((ext_vector_type(8)));

#define B_DIM   4
#define C_DIM   256
#define N_DIM   4096
#define TOPK_N  8
#define WAVES   8
#define WCOLS   (WAVES * 16)   // 128 columns per outer step
#define NBLK    (B_DIM * (N_DIM / 16))  // 1024 workgroups

// ------------------------------------------------------------------
// Kernel 1: inverse column norms  invn[b*N + i] = 1 / max(||feat[:,i]||, eps)
// ------------------------------------------------------------------
__global__ __launch_bounds__(256) void norm_kernel(const float* __restrict__ feats,
                                                   float* __restrict__ invn) {
    int gid = blockIdx.x * 256 + threadIdx.x;      // 0 .. 16383
    int b = gid >> 12;
    int i = gid & (N_DIM - 1);
    const float* p = feats + b * C_DIM * N_DIM + i;
    float s = 0.0f;
#pragma unroll 8
    for (int c = 0; c < C_DIM; ++c) {
        float v = p[c * N_DIM];
        s += v * v;
    }
    invn[gid] = 1.0f / fmaxf(sqrtf(s), 1e-12f);
}

// ------------------------------------------------------------------
// Kernel 2: fused  sim = N^T N  (f32 WMMA)  +  streaming top-8  +  loss partial
// One block (8 waves) per (batch, 16-row tile).
// ------------------------------------------------------------------
__global__ __launch_bounds__(256) void sim_topk_kernel(const float* __restrict__ feats,
                                                       const float* __restrict__ logits,
                                                       const float* __restrict__ invn,
                                                       float* __restrict__ partials) {
    const int bid  = blockIdx.x;
    const int b    = bid >> 8;          // batch
    const int I    = bid & 255;         // 16-row tile index
    const int tid  = threadIdx.x;
    const int wave = tid >> 5;
    const int lane = tid & 31;
    const int col16 = lane & 15;        // N (and A-row M) index inside tile
    const int khalf = lane >> 4;        // which K-pair this lane half owns

    __shared__ float ldsA[C_DIM * 16];          // A tile, [k][m], normalized (16 KB)
    __shared__ float ldsS[16 * WCOLS];          // sim tile strip [m][col]       (8 KB)
    __shared__ float ldsMV[16 * 16 * TOPK_N];   // merge values                  (8 KB)
    __shared__ int   ldsMI[16 * 16 * TOPK_N];   // merge indices                 (8 KB)
    __shared__ float rowsum[16];

    const float* fb  = feats + b * C_DIM * N_DIM;
    const float* inb = invn + b * N_DIM;
    const int I16 = I * 16;

    // ---- load + normalize A tile into LDS (coalesced) ----
#pragma unroll
    for (int e = 0; e < 16; ++e) {
        int f = e * 256 + tid;          // 0..4095
        int k = f >> 4;
        int m = f & 15;
        ldsA[f] = fb[k * N_DIM + I16 + m] * inb[I16 + m];
    }
    __syncthreads();

    // ---- per-thread streaming top-8 state (registers) ----
    const int r = tid >> 4;             // row this thread screens (0..15)
    const int s = tid & 15;             // column subset (0..15)
    float tval[TOPK_N];
    int   tidx[TOPK_N];
#pragma unroll
    for (int e = 0; e < TOPK_N; ++e) { tval[e] = -1e30f; tidx[e] = 0; }
    float mn = -1e30f;
    int   mnp = 0;

    for (int J0 = 0; J0 < N_DIM; J0 += WCOLS) {
        const int mycol = J0 + wave * 16 + col16;
        const float binv = inb[mycol];
        const float* bp = fb + mycol + (khalf * 2) * N_DIM;
        // hint: pull next column strip toward the caches
        if (J0 + WCOLS < N_DIM) __builtin_prefetch(bp + WCOLS, 0, 1);

        v8f acc = {};
#pragma unroll 4
        for (int kk = 0; kk < C_DIM; kk += 4) {
            const int kb = kk + khalf * 2;
            v2f a, bb;
            a.x  = ldsA[kb * 16 + col16];
            a.y  = ldsA[(kb + 1) * 16 + col16];
            bb.x = bp[kk * N_DIM] * binv;
            bb.y = bp[kk * N_DIM + N_DIM] * binv;
            acc = __builtin_amdgcn_wmma_f32_16x16x4_f32(
                /*neg_a=*/false, a, /*neg_b=*/false, bb,
                /*c_mod=*/(short)0, acc, /*reuse_a=*/false, /*reuse_b=*/false);
        }

        // ---- spill 16x16 C tile to LDS strip ----
#pragma unroll
        for (int v = 0; v < 8; ++v) {
            int m = v + khalf * 8;
            ldsS[m * WCOLS + wave * 16 + col16] = acc[v];
        }
        __syncthreads();

        // ---- screen 8 candidates per thread into its top-8 list ----
#pragma unroll
        for (int c = 0; c < WCOLS; c += 16) {
            float val = ldsS[r * WCOLS + c + s];
            if (val > mn) {
                tval[mnp] = val;
                tidx[mnp] = J0 + c + s;
                mn = tval[0]; mnp = 0;
#pragma unroll
                for (int e = 1; e < TOPK_N; ++e)
                    if (tval[e] < mn) { mn = tval[e]; mnp = e; }
            }
        }
        __syncthreads();
    }

    // ---- merge 16 partial lists per row ----
#pragma unroll
    for (int e = 0; e < TOPK_N; ++e) {
        ldsMV[r * 128 + s * TOPK_N + e] = tval[e];
        ldsMI[r * 128 + s * TOPK_N + e] = tidx[e];
    }
    __syncthreads();

    if (tid < 16) {
        float fv[TOPK_N]; int fi[TOPK_N];
#pragma unroll
        for (int e = 0; e < TOPK_N; ++e) { fv[e] = -1e30f; fi[e] = 0; }
        float m2 = -1e30f; int m2p = 0;
        for (int q = 0; q < 128; ++q) {
            float val = ldsMV[tid * 128 + q];
            if (val > m2) {
                fv[m2p] = val;
                fi[m2p] = ldsMI[tid * 128 + q];
                m2 = fv[0]; m2p = 0;
#pragma unroll
                for (int e = 1; e < TOPK_N; ++e)
                    if (fv[e] < m2) { m2 = fv[e]; m2p = e; }
            }
        }
        // ---- loss contribution for this row ----
        const float* lg = logits + b * N_DIM;
        int gi = I16 + tid;
        float anchor = 1.0f / (1.0f + expf(-lg[gi]));
        float ssum = 0.0f;
#pragma unroll
        for (int e = 0; e < TOPK_N; ++e) {
            float pe = 1.0f / (1.0f + expf(-lg[fi[e]]));
            ssum += fabsf(anchor - pe);
        }
        rowsum[tid] = ssum;
    }
    __syncthreads();

    if (tid == 0) {
        float t = 0.0f;
#pragma unroll
        for (int e = 0; e < 16; ++e) t += rowsum[e];
        partials[bid] = t;      // one deterministic partial per block
    }
}

// ------------------------------------------------------------------
// Kernel 3: deterministic serial reduction of 1024 partials -> scalar loss
// ------------------------------------------------------------------
__global__ void finalize_kernel(const float* __restrict__ partials,
                                float* __restrict__ out) {
    if (blockIdx.x == 0 && threadIdx.x == 0) {
        float t = 0.0f;
        for (int i = 0; i < NBLK; ++i) t += partials[i];
        out[0] = t * (1.0f / (float)(B_DIM * N_DIM * TOPK_N));
    }
}

// ------------------------------------------------------------------
extern "C" void kernel_launch(void* const* d_in, const int* in_sizes, int n_in,
                              void* d_out, int out_size, void* d_ws, size_t ws_size,
                              hipStream_t stream) {
    (void)in_sizes; (void)n_in; (void)out_size; (void)ws_size;
    const float* feats  = (const float*)d_in[0];   // (4,256,64,64) f32
    const float* logits = (const float*)d_in[1];   // (4,1,64,64)   f32
    float* invn     = (float*)d_ws;                // B*N floats
    float* partials = invn + B_DIM * N_DIM;        // NBLK floats

    norm_kernel<<<(B_DIM * N_DIM) / 256, 256, 0, stream>>>(feats, invn);
    sim_topk_kernel<<<NBLK, 256, 0, stream>>>(feats, logits, invn, partials);
    finalize_kernel<<<1, 32, 0, stream>>>(partials, (float*)d_out);
}